// DirectionalBSplineGrid_46505905881446
// MI455X (gfx1250) — compile-verified
//
#include <hip/hip_runtime.h>
#include <hip/hip_bf16.h>

#define N_ALPHA 12
#define N_DEPTH 10

__device__ __forceinline__ float clampf(float x, float lo, float hi) {
  return fminf(fmaxf(x, lo), hi);
}

// ---- asin poly (A&S 4.4.46, |err|<=2e-8) with the reference's
// normalization  a_idx = 11*(asin(y)+pi/2)/(pi+1e-8)  folded into the
// coefficients:  a_idx = y>=0 ? C1A - sqrt(1-t)*P(t) : sqrt(1-t)*P(t).
#define PI_D 3.14159265358979323846
#define ASCL (11.0 / (PI_D + 1e-8))               // 11/(pi+eps)

__device__ __forceinline__ float alpha_index(float y) {
  const float C1A = (float)(PI_D * ASCL);         // ~10.99999996
  const float P0 = (float)(1.5707963050 * ASCL);
  const float P1 = (float)(-0.2145988016 * ASCL);
  const float P2 = (float)(0.0889789874 * ASCL);
  const float P3 = (float)(-0.0501743046 * ASCL);
  const float P4 = (float)(0.0308918810 * ASCL);
  const float P5 = (float)(-0.0170881256 * ASCL);
  const float P6 = (float)(0.0066700901 * ASCL);
  const float P7 = (float)(-0.0012624911 * ASCL);
  float t = fminf(fabsf(y), 1.0f);
  float s = sqrtf(1.0f - t);
  float p = fmaf(P7, t, P6);
  p = fmaf(p, t, P5);
  p = fmaf(p, t, P4);
  p = fmaf(p, t, P3);
  p = fmaf(p, t, P2);
  p = fmaf(p, t, P1);
  p = fmaf(p, t, P0);
  float q = s * p;                        // 11*(pi/2 - asin(t))/(pi+eps)
  float ai = (y >= 0.0f) ? (C1A - q) : q;
  return clampf(ai, 0.0f, 11.0f);
}

// Cubic uniform B-spline basis from u, u^2, u^3 (all FMA-shaped, no (1-u)):
//   b0 = 1/6 - u/2 + u^2/2 - u^3/6
//   b1 = 2/3 - u^2 + u^3/2
//   b2 = 1/6 + u/2 + u^2/2 - u^3/2
//   b3 = u^3/6
__device__ __forceinline__ void basis4(float u, float& b0, float& b1,
                                       float& b2, float& b3) {
  const float SIXTH = 1.0f / 6.0f;
  float u2 = u * u;
  float u3 = u2 * u;
  b0 = fmaf(u3, -SIXTH, fmaf(u2, 0.5f, fmaf(u, -0.5f, SIXTH)));
  b1 = fmaf(u3, 0.5f, fmaf(u2, -1.0f, 2.0f / 3.0f));
  b2 = fmaf(u3, -0.5f, fmaf(u2, 0.5f, fmaf(u, 0.5f, SIXTH)));
  b3 = u3 * SIXTH;
}

// Evaluate the clamped bicubic B-spline for one pixel; cp = 12x10 table (LDS).
__device__ __forceinline__ float bspline_eval(float ray_y, float logd,
                                              const float* __restrict__ cp) {
  const float KD  = (float)(9.0 / (8.0 + 1e-8));   // 9/(8+eps)
  const float KD3 = (float)(27.0 / (8.0 + 1e-8));  // 3*KD

  float ai = alpha_index(ray_y);
  float di = clampf(fmaf(logd, KD, KD3), 0.0f, 9.0f);

  float fa = floorf(ai), fd = floorf(di);
  float ua = ai - fa,    ud = di - fd;
  int a0 = min(max((int)fa - 1, 0), N_ALPHA - 4);  // [0, 8]
  int d0 = min(max((int)fd - 1, 0), N_DEPTH - 4);  // [0, 6]

  float ab0, ab1, ab2, ab3, db0, db1, db2, db3;
  basis4(ua, ab0, ab1, ab2, ab3);
  basis4(ud, db0, db1, db2, db3);

  // 4x4 gather; indices provably in [0,119]. Plain scalar FMA chains —
  // VOPD dual-issue pairs these.
  const float* r = cp + (a0 * N_DEPTH + d0);
  float s0 = fmaf(r[3], db3, fmaf(r[2], db2, fmaf(r[1], db1, r[0] * db0)));
  r += N_DEPTH;
  float s1 = fmaf(r[3], db3, fmaf(r[2], db2, fmaf(r[1], db1, r[0] * db0)));
  r += N_DEPTH;
  float s2 = fmaf(r[3], db3, fmaf(r[2], db2, fmaf(r[1], db1, r[0] * db0)));
  r += N_DEPTH;
  float s3 = fmaf(r[3], db3, fmaf(r[2], db2, fmaf(r[1], db1, r[0] * db0)));

  float res = fmaf(ab3, s3, fmaf(ab2, s2, fmaf(ab1, s1, ab0 * s0)));
  return clampf(res, -0.5f, 0.5f);
}

// Main kernel: 4 pixels per thread, fully vectorized b128 global traffic.
__global__ void __launch_bounds__(256)
spline4_kernel(const float* __restrict__ rd,   // ray_dirs (..,3)
               const float* __restrict__ ld,   // log_depth
               const float* __restrict__ cp,   // control_points (120)
               float* __restrict__ out, int n4) {
  __shared__ float s_cp[128];  // 120 used

  const int tid = threadIdx.x;

  // ---- stage control table into LDS -------------------------------------
  // CDNA5 async global->LDS DMA, wait ASYNCcnt, then a plain ds_store of the
  // same values (issued AFTER the wait) so correctness never depends on the
  // async op's semantics — any stray async writes inside s_cp get healed.
  if (tid < 120) {
#if defined(__gfx1250__)
    const float* g = cp + tid;
    unsigned lds_off = (unsigned)(unsigned long long)&s_cp[tid];
    asm volatile("global_load_async_to_lds_b32 %0, %1, off"
                 :: "v"(lds_off), "v"(g)
                 : "memory");
#endif
  }
#if defined(__gfx1250__)
  asm volatile("s_wait_asynccnt 0x0" ::: "memory");
#endif
  if (tid < 120) {
    s_cp[tid] = cp[tid];   // authoritative copy
  }
  __syncthreads();

  const int t = blockIdx.x * blockDim.x + tid;
  if (t >= n4) return;

  // 4 pixels -> ray y components at flat elements 12t+{1,4,7,10}
  const float4* rd4 = reinterpret_cast<const float4*>(rd);
  float4 r0 = rd4[3 * t + 0];
  float4 r1 = rd4[3 * t + 1];
  float4 r2 = rd4[3 * t + 2];
  float4 l4 = reinterpret_cast<const float4*>(ld)[t];

  float4 o;
  o.x = bspline_eval(r0.y, l4.x, s_cp);
  o.y = bspline_eval(r1.x, l4.y, s_cp);
  o.z = bspline_eval(r1.w, l4.z, s_cp);
  o.w = bspline_eval(r2.z, l4.w, s_cp);

  reinterpret_cast<float4*>(out)[t] = o;
}

// Scalar tail (never launched for the fixed 4x1024x2048 shape).
__global__ void spline1_kernel(const float* __restrict__ rd,
                               const float* __restrict__ ld,
                               const float* __restrict__ cp,
                               float* __restrict__ out, int start, int total) {
  int i = start + (int)(blockIdx.x * blockDim.x + threadIdx.x);
  if (i >= total) return;
  out[i] = bspline_eval(rd[3 * i + 1], ld[i], cp);
}

extern "C" void kernel_launch(void* const* d_in, const int* in_sizes, int n_in,
                              void* d_out, int out_size, void* d_ws, size_t ws_size,
                              hipStream_t stream) {
  (void)n_in; (void)d_ws; (void)ws_size; (void)out_size;
  const float* rd = (const float*)d_in[0];   // ray_dirs  (B,H,W,3)
  const float* ld = (const float*)d_in[1];   // log_depth (B,H,W)
  const float* cp = (const float*)d_in[2];   // control_points (12,10)
  float* out = (float*)d_out;

  const int total = in_sizes[1];             // B*H*W
  const int n4 = total >> 2;
  if (n4 > 0) {
    dim3 grid((unsigned)((n4 + 255) / 256));
    spline4_kernel<<<grid, 256, 0, stream>>>(rd, ld, cp, out, n4);
  }
  const int tail = total & 3;
  if (tail) {
    spline1_kernel<<<1, 64, 0, stream>>>(rd, ld, cp, out, n4 << 2, total);
  }
}